// GatedLayer_25512105738336
// MI455X (gfx1250) — compile-verified
//
#include <hip/hip_runtime.h>
#include <cstdint>
#include <cstddef>

// ---------------------------------------------------------------------------
// CDNA5 async global->LDS helpers (gfx1250).
// Inline asm (assembler-validated) to avoid clang builtin signature mismatches:
// GLOBAL_LOAD_ASYNC_TO_LDS_B128: vdst = LDS byte address VGPR, vaddr = 64-bit
// global address VGPR pair, "off" = no SADDR. Tracked by ASYNCcnt.
// ---------------------------------------------------------------------------
#ifndef __has_builtin
#define __has_builtin(x) 0
#endif

__device__ __forceinline__ void async_copy_b128_to_lds(const void* g, void* l) {
  unsigned lds = (unsigned)(unsigned long long)l;  // low 32 bits = LDS byte addr
  asm volatile("global_load_async_to_lds_b128 %0, %1, off"
               :: "v"(lds), "v"(g) : "memory");
}

__device__ __forceinline__ void wait_async0() {
#if __has_builtin(__builtin_amdgcn_s_wait_asynccnt)
  __builtin_amdgcn_s_wait_asynccnt(0);
#else
  asm volatile("s_wait_asynccnt 0x0" ::: "memory");
#endif
}

// Relaxed agent-scope f32 atomic add -> global_atomic_add_f32 (no CAS loop).
__device__ __forceinline__ void atomicAddF(float* p, float v) {
  __hip_atomic_fetch_add(p, v, __ATOMIC_RELAXED, __HIP_MEMORY_SCOPE_AGENT);
}

// ---------------------------------------------------------------------------
// Kernels
// ---------------------------------------------------------------------------

__global__ void k_zero(float4* p, long long n4) {
  long long i = (long long)blockIdx.x * blockDim.x + threadIdx.x;
  long long stride = (long long)gridDim.x * blockDim.x;
  for (; i < n4; i += stride) p[i] = make_float4(0.f, 0.f, 0.f, 0.f);
}

// argmax over 32 logits per node, first-max-wins (matches jnp.argmax ties)
__global__ void k_argmax(const float* __restrict__ logits, int* __restrict__ pred, int N) {
  int i = blockIdx.x * blockDim.x + threadIdx.x;
  if (i >= N) return;
  const float4* row = (const float4*)(logits + (long long)i * 32);
  float best = -3.402823466e+38f;
  int bi = 0;
#pragma unroll
  for (int j = 0; j < 8; ++j) {
    float4 v = row[j];
    if (v.x > best) { best = v.x; bi = j * 4 + 0; }
    if (v.y > best) { best = v.y; bi = j * 4 + 1; }
    if (v.z > best) { best = v.z; bi = j * 4 + 2; }
    if (v.w > best) { best = v.w; bi = j * 4 + 3; }
  }
  pred[i] = bi;
}

// Per-edge: degree, class histogram, prediction-match. Edge indices for full
// 256-edge tiles are staged into LDS with async global->LDS b128 copies.
__global__ void __launch_bounds__(256)
k_edge(const int* __restrict__ src, const int* __restrict__ dst,
       const int* __restrict__ pred,
       int* __restrict__ deg, int* __restrict__ counts,
       float* __restrict__ f1sum, long long E) {
  __shared__ int s_s[256];
  __shared__ int s_d[256];
  long long base = (long long)blockIdx.x * 256;
  int t = threadIdx.x;
  int s = 0, d = 0;
  bool valid;
  if (base + 256 <= E) {
    if (t < 64) {                       // waves 0-1: stage src chunk (1 KB)
      async_copy_b128_to_lds(src + base + (long long)t * 4, &s_s[t * 4]);
    } else if (t < 128) {               // waves 2-3: stage dst chunk (1 KB)
      int q = t - 64;
      async_copy_b128_to_lds(dst + base + (long long)q * 4, &s_d[q * 4]);
    }
    wait_async0();
    __syncthreads();
    s = s_s[t];
    d = s_d[t];
    valid = true;
  } else {                              // partial tail tile: direct loads
    long long e = base + t;
    valid = e < E;
    if (valid) { s = src[e]; d = dst[e]; }
  }
  if (valid) {
    int ps = pred[s];
    int pd = pred[d];
    atomicAdd(&deg[d], 1);
    atomicAdd(&counts[(long long)d * 32 + ps], 1);
    if (ps == pd) atomicAddF(&f1sum[d], 1.0f);
  }
}

// agg[dst] += feats[src]; 24 threads per edge, one float4 each (coalesced
// 384B row gathers out of L2), 4 hardware f32 atomic adds per thread.
__global__ void k_edge_feat(const int* __restrict__ src, const int* __restrict__ dst,
                            const float* __restrict__ feats, float* __restrict__ agg,
                            long long total) {
  long long t = (long long)blockIdx.x * blockDim.x + threadIdx.x;
  if (t >= total) return;
  int e = (int)(t / 24);
  int j = (int)(t % 24);
  int s = src[e];
  int d = dst[e];
  const float4 v = ((const float4*)feats)[(long long)s * 24 + j];
  float* a = agg + (long long)d * 96 + (long long)j * 4;
  atomicAddF(a + 0, v.x);
  atomicAddF(a + 1, v.y);
  atomicAddF(a + 2, v.z);
  atomicAddF(a + 3, v.w);
}

// Per-node f1 (match fraction) and f2 (neighbor-class entropy with 1e-5 clip),
// plus block-reduced sum / sum-of-squares for both layernorms.
__global__ void __launch_bounds__(256)
k_node_stats(const int* __restrict__ deg, const float* __restrict__ f1sum,
             const int* __restrict__ counts,
             float* __restrict__ f1, float* __restrict__ f2,
             float* __restrict__ stats, int N) {
  int i = blockIdx.x * blockDim.x + threadIdx.x;
  float a = 0.f, b = 0.f, c = 0.f, d = 0.f;
  if (i < N) {
    int dr = deg[i];
    float degs = (float)(dr > 1 ? dr : 1);
    float inv = 1.0f / degs;
    float f1v = 0.f, f2v = 0.f;
    if (dr > 0) {
      f1v = f1sum[i] * inv;
      const int* row = counts + (long long)i * 32;
      float s = 0.f;
#pragma unroll
      for (int k = 0; k < 32; ++k) {
        float p = fmaxf((float)row[k] * inv, 1e-5f);
        s += p * logf(p);
      }
      f2v = -s;
    }
    f1[i] = f1v;
    f2[i] = f2v;
    a = f1v; b = f1v * f1v; c = f2v; d = f2v * f2v;
  }
  __shared__ float r[4][256];
  int t = threadIdx.x;
  r[0][t] = a; r[1][t] = b; r[2][t] = c; r[3][t] = d;
  __syncthreads();
  for (int off = 128; off > 0; off >>= 1) {
    if (t < off) {
      r[0][t] += r[0][t + off];
      r[1][t] += r[1][t + off];
      r[2][t] += r[2][t + off];
      r[3][t] += r[3][t + off];
    }
    __syncthreads();
  }
  if (t == 0) {
    atomicAddF(&stats[0], r[0][0]);
    atomicAddF(&stats[1], r[1][0]);
    atomicAddF(&stats[2], r[2][0]);
    atomicAddF(&stats[3], r[3][0]);
  }
}

// Fused epilogue: layernorm both stats, gate, min with old_z, degree-normalized
// residual add. 24 threads/node, float4 in/out. z written by the j==0 thread.
__global__ void k_final(const float* __restrict__ feats, const float* __restrict__ old_z,
                        const float* __restrict__ tau1, const float* __restrict__ tau2,
                        const int* __restrict__ deg, const float* __restrict__ agg,
                        const float* __restrict__ f1, const float* __restrict__ f2,
                        const float* __restrict__ stats,
                        float* __restrict__ out_h, float* __restrict__ out_z, int N) {
  long long t = (long long)blockIdx.x * blockDim.x + threadIdx.x;
  int i = (int)(t / 24);
  int j = (int)(t % 24);
  if (i >= N) return;
  float invN = 1.0f / (float)N;
  float m1 = stats[0] * invN;
  float v1 = fmaxf(stats[1] * invN - m1 * m1, 0.0f);
  float r1 = 1.0f / sqrtf(v1 + 1e-5f);
  float m2 = stats[2] * invN;
  float v2 = fmaxf(stats[3] * invN - m2 * m2, 0.0f);
  float r2 = 1.0f / sqrtf(v2 + 1e-5f);
  float nf1 = (f1[i] - m1) * r1;
  float nf2 = (f2[i] - m2) * r2;
  // sigmoid(-(x - tau)) == 1 / (1 + exp(x - tau))
  float z = 1.0f / (1.0f + expf(nf1 - tau1[0]));
  z *= 1.0f / (1.0f + expf(nf2 - tau2[0]));
  float gate = fminf(old_z[i], z);
  int dr = deg[i];
  float nrm = 1.0f / sqrtf((float)(dr > 1 ? dr : 1));
  float g = gate * nrm;
  const float4 fv = ((const float4*)feats)[(long long)i * 24 + j];
  const float4 av = ((const float4*)agg)[(long long)i * 24 + j];
  float4 o;
  o.x = fv.x + g * av.x;
  o.y = fv.y + g * av.y;
  o.z = fv.z + g * av.z;
  o.w = fv.w + g * av.w;
  ((float4*)out_h)[(long long)i * 24 + j] = o;
  if (j == 0) out_z[i] = z;
}

// ---------------------------------------------------------------------------
// Launcher
// ---------------------------------------------------------------------------
extern "C" void kernel_launch(void* const* d_in, const int* in_sizes, int n_in,
                              void* d_out, int out_size, void* d_ws, size_t ws_size,
                              hipStream_t stream) {
  (void)n_in; (void)out_size; (void)ws_size;
  const float* feats  = (const float*)d_in[0];   // [N,96]
  const float* logits = (const float*)d_in[1];   // [N,32]
  const float* old_z  = (const float*)d_in[2];   // [N]
  const float* tau1   = (const float*)d_in[3];   // [1]
  const float* tau2   = (const float*)d_in[4];   // [1]
  const int*   src    = (const int*)d_in[5];     // [E]
  const int*   dst    = (const int*)d_in[6];     // [E]
  const int N = in_sizes[2];
  const long long E = in_sizes[5];

  // Workspace carve-out (16B aligned). Zeroed region first.
  char* ws = (char*)d_ws;
  size_t off = 0;
  auto carve = [&](size_t bytes) -> void* {
    void* p = ws + off;
    off = (off + bytes + 15) & ~(size_t)15;
    return p;
  };
  float* agg    = (float*)carve((size_t)N * 96 * sizeof(float));
  int*   counts = (int*)  carve((size_t)N * 32 * sizeof(int));
  int*   deg    = (int*)  carve((size_t)N * sizeof(int));
  float* f1sum  = (float*)carve((size_t)N * sizeof(float));
  float* stats  = (float*)carve(4 * sizeof(float));
  size_t zero_bytes = off;
  int*   pred = (int*)  carve((size_t)N * sizeof(int));
  float* f1   = (float*)carve((size_t)N * sizeof(float));
  float* f2   = (float*)carve((size_t)N * sizeof(float));

  float* out_h = (float*)d_out;
  float* out_z = out_h + (size_t)N * 96;

  long long n4 = (long long)(zero_bytes / 16);
  int zgrid = (int)((n4 + 255) / 256);
  if (zgrid > 8192) zgrid = 8192;
  k_zero<<<zgrid, 256, 0, stream>>>((float4*)d_ws, n4);

  k_argmax<<<(N + 255) / 256, 256, 0, stream>>>(logits, pred, N);

  k_edge<<<(int)((E + 255) / 256), 256, 0, stream>>>(src, dst, pred, deg, counts, f1sum, E);

  long long total = E * 24;
  k_edge_feat<<<(int)((total + 255) / 256), 256, 0, stream>>>(src, dst, feats, agg, total);

  k_node_stats<<<(N + 255) / 256, 256, 0, stream>>>(deg, f1sum, counts, f1, f2, stats, N);

  long long nt = (long long)N * 24;
  k_final<<<(int)((nt + 255) / 256), 256, 0, stream>>>(feats, old_z, tau1, tau2, deg, agg,
                                                       f1, f2, stats, out_h, out_z, N);
}